// InvResMLP_64510408786138
// MI455X (gfx1250) — compile-verified
//
#include <hip/hip_runtime.h>
#include <cfloat>

typedef float v2f __attribute__((ext_vector_type(2)));
typedef float v8f __attribute__((ext_vector_type(8)));

#define BB   4
#define NN   16384
#define KK   32
#define CC   64
#define MIDC 256
#define EPSF 1e-5f

__device__ __forceinline__ v8f wmma_f32(v2f a, v2f b, v8f c) {
    // V_WMMA_F32_16X16X4_F32: (neg_a, A, neg_b, B, c_mod, C, reuse_a, reuse_b)
    return __builtin_amdgcn_wmma_f32_16x16x4_f32(false, a, false, b, (short)0, c, false, false);
}

__global__ void k_zero(float* __restrict__ p, int n) {
    int i = blockIdx.x * blockDim.x + threadIdx.x;
    if (i < n) p[i] = 0.f;
}

// f [B,C,N] -> f_t [B,N,C], LDS-tiled transpose, coalesced both ways.
__global__ void k_transpose(const float* __restrict__ f, float* __restrict__ ft) {
    __shared__ float tile[32][33];
    int b = blockIdx.z, c0 = blockIdx.y * 32, n0 = blockIdx.x * 32;
    int tx = threadIdx.x, ty = threadIdx.y;
    for (int i = 0; i < 4; i++) {
        int c = c0 + ty + i * 8;
        tile[ty + i * 8][tx] = f[((size_t)b * CC + c) * NN + n0 + tx];
    }
    __syncthreads();
    for (int i = 0; i < 4; i++) {
        int n = n0 + ty + i * 8;
        ft[((size_t)b * NN + n) * CC + c0 + tx] = tile[tx][ty + i * 8];
    }
}

// One 16-row (k) x 64-col tile of GEMM0: feats'=[fj(64)|dp(3)|0] x w0'^T.
// A fragment per f32 16x16x4 layout: lane -> row m=lane%16, k-pair selected by lane/16.
__device__ __forceinline__ void run_tile0(const float* __restrict__ frow,
                                          float dx, float dy, float dz,
                                          int m, int hf, const float* WT, v8f acc[4]) {
    for (int j = 0; j < 16; j++) {            // fj part of inner dim
        int kk = 4 * j + 2 * hf;
        v2f A = {frow[kk], frow[kk + 1]};
#pragma unroll
        for (int ct = 0; ct < 4; ct++) {
            int col = ct * 16 + m;
            v2f Bv = {WT[col * 68 + kk], WT[col * 68 + kk + 1]};
            acc[ct] = wmma_f32(A, Bv, acc[ct]);
        }
    }
    {                                          // tail: dp + zero pad (kk = 64..67)
        int kk = 64 + 2 * hf;
        v2f A = {hf ? dz : dx, hf ? 0.f : dy};
#pragma unroll
        for (int ct = 0; ct < 4; ct++) {
            int col = ct * 16 + m;
            v2f Bv = {WT[col * 68 + kk], WT[col * 68 + kk + 1]};
            acc[ct] = wmma_f32(A, Bv, acc[ct]);
        }
    }
}

// Stage 0: per point n, gather K=32 neighbors, h = feats' @ w0'^T (67->64),
// reduce max/min over K and channel sum/sumsq for BN0 stats.
// One wave per point; the two 16-row k-tiles run SEQUENTIALLY (halves live VGPRs).
__global__ void __launch_bounds__(128, 1)
k_stage0(const float* __restrict__ p, const float* __restrict__ ft,
         const int* __restrict__ idx, const float* __restrict__ w0,
         float* __restrict__ hmax, float* __restrict__ hmin,
         float* __restrict__ stats0) {
    __shared__ float WT[CC * 68];   // WT[col][kk], reordered+padded w0, stride 68 (bank-safe)
    __shared__ float lsum[CC], lsq[CC];
    int tid = threadIdx.x;
    if (tid < CC) { lsum[tid] = 0.f; lsq[tid] = 0.f; }
    for (int i = tid; i < CC * 68; i += 128) {
        int col = i / 68, kk = i % 68;
        float v = 0.f;
        if (kk < 64)      v = w0[col * 67 + 3 + kk];    // fj part
        else if (kk < 67) v = w0[col * 67 + (kk - 64)]; // dp part
        WT[i] = v;
    }
    __syncthreads();

    int wave = tid >> 5, lane = tid & 31;
    int m = lane & 15, hf = lane >> 4;
    int gw = blockIdx.x * 4 + wave;
    int b = gw >> 14, n = gw & (NN - 1);
    size_t row = (size_t)b * NN + n;

    int nbr0 = idx[row * KK + m];
    int nbr1 = idx[row * KK + 16 + m];
    const float* pc = p + row * 3;
    float cx = pc[0], cy = pc[1], cz = pc[2];
    const float* p0 = p + ((size_t)b * NN + nbr0) * 3;
    const float* p1 = p + ((size_t)b * NN + nbr1) * 3;
    float d0x = p0[0] - cx, d0y = p0[1] - cy, d0z = p0[2] - cz;
    float d1x = p1[0] - cx, d1y = p1[1] - cy, d1z = p1[2] - cz;
    const float* f0 = ft + ((size_t)b * NN + nbr0) * CC;
    const float* f1 = ft + ((size_t)b * NN + nbr1) * CC;

    v8f zero = {0.f, 0.f, 0.f, 0.f, 0.f, 0.f, 0.f, 0.f};
    float ps[4], pq[4], pmx[4], pmn[4];
#pragma unroll
    for (int ct = 0; ct < 4; ct++) { ps[ct] = 0.f; pq[ct] = 0.f; pmx[ct] = -FLT_MAX; pmn[ct] = FLT_MAX; }

    v8f acc[4] = {zero, zero, zero, zero};
    run_tile0(f0, d0x, d0y, d0z, m, hf, WT, acc);       // k = 0..15
#pragma unroll
    for (int ct = 0; ct < 4; ct++) {
#pragma unroll
        for (int jj = 0; jj < 8; jj++) {
            float x = acc[ct][jj];
            ps[ct] += x; pq[ct] += x * x;
            pmx[ct] = fmaxf(pmx[ct], x); pmn[ct] = fminf(pmn[ct], x);
        }
        acc[ct] = zero;
    }
    run_tile0(f1, d1x, d1y, d1z, m, hf, WT, acc);       // k = 16..31
#pragma unroll
    for (int ct = 0; ct < 4; ct++) {
#pragma unroll
        for (int jj = 0; jj < 8; jj++) {
            float x = acc[ct][jj];
            ps[ct] += x; pq[ct] += x * x;
            pmx[ct] = fmaxf(pmx[ct], x); pmn[ct] = fminf(pmn[ct], x);
        }
    }

#pragma unroll
    for (int ct = 0; ct < 4; ct++) {
        float s = ps[ct] + __shfl_xor(ps[ct], 16, 32);
        float q = pq[ct] + __shfl_xor(pq[ct], 16, 32);
        float mx = fmaxf(pmx[ct], __shfl_xor(pmx[ct], 16, 32));
        float mn = fminf(pmn[ct], __shfl_xor(pmn[ct], 16, 32));
        if (lane < 16) {
            int c = ct * 16 + m;
            atomicAdd(&lsum[c], s);
            atomicAdd(&lsq[c], q);
            hmax[row * CC + c] = mx;
            hmin[row * CC + c] = mn;
        }
    }
    __syncthreads();
    if (tid < CC) {
        atomicAdd(&stats0[tid], lsum[tid]);
        atomicAdd(&stats0[CC + tid], lsq[tid]);
    }
}

__global__ void k_finalize(const float* __restrict__ stats, const float* __restrict__ g,
                           const float* __restrict__ be, float* __restrict__ sb,
                           float inv_count, int nch) {
    int t = blockIdx.x * blockDim.x + threadIdx.x;
    if (t < nch) {
        float mean = stats[t] * inv_count;
        float var  = stats[nch + t] * inv_count - mean * mean;
        float inv  = rsqrtf(fmaxf(var, 0.f) + EPSF);
        float sc   = g[t] * inv;
        sb[t]       = sc;
        sb[nch + t] = be[t] - mean * sc;
    }
}

// Stage 1: fl = relu(bn0(max_k h))  (sign-aware via hmax/hmin), h1 = fl @ w1^T.
// Block = 16 rows; each wave handles 64 of 256 output channels; A tile in LDS (stride 68).
__global__ void __launch_bounds__(128, 1)
k_stage1(const float* __restrict__ hmax, const float* __restrict__ hmin,
         const float* __restrict__ sb0, const float* __restrict__ w1,
         float* __restrict__ h1, float* __restrict__ stats1) {
    __shared__ float flT[16 * 68];
    int tid = threadIdx.x;
    int rowbase = blockIdx.x * 16;
    for (int i = tid; i < 16 * CC; i += 128) {
        int r = i >> 6, c = i & 63;
        float s = sb0[c];
        size_t o = ((size_t)(rowbase + r)) * CC + c;
        float h = (s >= 0.f) ? hmax[o] : hmin[o];
        flT[r * 68 + c] = fmaxf(s * h + sb0[CC + c], 0.f);
    }
    __syncthreads();

    int wave = tid >> 5, lane = tid & 31, m = lane & 15, hf = lane >> 4;
    int colbase = wave * 64;
    v8f zero = {0.f, 0.f, 0.f, 0.f, 0.f, 0.f, 0.f, 0.f};
    v8f acc[4] = {zero, zero, zero, zero};

    for (int j = 0; j < 16; j++) {
        int kk = 4 * j + 2 * hf;
        v2f A = {flT[m * 68 + kk], flT[m * 68 + kk + 1]};
#pragma unroll
        for (int ct = 0; ct < 4; ct++) {
            int col = colbase + ct * 16 + m;
            v2f Bv = {w1[col * CC + kk], w1[col * CC + kk + 1]};
            acc[ct] = wmma_f32(A, Bv, acc[ct]);
        }
    }
#pragma unroll
    for (int ct = 0; ct < 4; ct++) {
        int col = colbase + ct * 16 + m;
        float s = 0.f, q = 0.f;
#pragma unroll
        for (int jj = 0; jj < 8; jj++) {
            float x = acc[ct][jj]; s += x; q += x * x;
            int r = rowbase + hf * 8 + jj;
            h1[(size_t)r * MIDC + col] = x;
        }
        s += __shfl_xor(s, 16, 32);
        q += __shfl_xor(q, 16, 32);
        if (lane < 16) {
            atomicAdd(&stats1[col], s);
            atomicAdd(&stats1[MIDC + col], q);
        }
    }
}

// Stage 2: h2 = relu(bn1(h1)) @ w2^T. Wave = 16-row tile, all 64 out cols, 64 k-steps.
__global__ void __launch_bounds__(128, 1)
k_stage2(const float* __restrict__ h1, const float* __restrict__ sb1,
         const float* __restrict__ w2, float* __restrict__ h2,
         float* __restrict__ stats2) {
    __shared__ float s1[MIDC], be1[MIDC];
    int tid = threadIdx.x;
    for (int i = tid; i < MIDC; i += 128) { s1[i] = sb1[i]; be1[i] = sb1[MIDC + i]; }
    __syncthreads();

    int wave = tid >> 5, lane = tid & 31, m = lane & 15, hf = lane >> 4;
    int rowbase = (blockIdx.x * 4 + wave) * 16;
    const float* hrow = h1 + (size_t)(rowbase + m) * MIDC;
    v8f zero = {0.f, 0.f, 0.f, 0.f, 0.f, 0.f, 0.f, 0.f};
    v8f acc[4] = {zero, zero, zero, zero};

    for (int j = 0; j < 64; j++) {
        int kk = 4 * j + 2 * hf;
        float r0 = hrow[kk], r1 = hrow[kk + 1];
        v2f A = {fmaxf(s1[kk] * r0 + be1[kk], 0.f),
                 fmaxf(s1[kk + 1] * r1 + be1[kk + 1], 0.f)};
#pragma unroll
        for (int ct = 0; ct < 4; ct++) {
            int col = ct * 16 + m;
            v2f Bv = {w2[col * MIDC + kk], w2[col * MIDC + kk + 1]};
            acc[ct] = wmma_f32(A, Bv, acc[ct]);
        }
    }
#pragma unroll
    for (int ct = 0; ct < 4; ct++) {
        int col = ct * 16 + m;
        float s = 0.f, q = 0.f;
#pragma unroll
        for (int jj = 0; jj < 8; jj++) {
            float x = acc[ct][jj]; s += x; q += x * x;
            int r = rowbase + hf * 8 + jj;
            h2[(size_t)r * CC + col] = x;
        }
        s += __shfl_xor(s, 16, 32);
        q += __shfl_xor(q, 16, 32);
        if (lane < 16) {
            atomicAdd(&stats2[col], s);
            atomicAdd(&stats2[CC + col], q);
        }
    }
}

// out[b,c,n] = relu(bn2(h2)[b,n,c] + f_t[b,n,c]), transposed store.
__global__ void k_out(const float* __restrict__ h2, const float* __restrict__ ft,
                      const float* __restrict__ sb2, float* __restrict__ out) {
    __shared__ float tile[32][33];
    int b = blockIdx.z, c0 = blockIdx.y * 32, n0 = blockIdx.x * 32;
    int tx = threadIdx.x, ty = threadIdx.y;
    int c = c0 + tx;
    float sc = sb2[c], bi = sb2[CC + c];
    for (int i = 0; i < 4; i++) {
        int n = n0 + ty + i * 8;
        size_t o = ((size_t)b * NN + n) * CC + c;
        tile[tx][ty + i * 8] = fmaxf(sc * h2[o] + bi + ft[o], 0.f);
    }
    __syncthreads();
    for (int i = 0; i < 4; i++) {
        int cc2 = c0 + ty + i * 8;
        out[((size_t)b * CC + cc2) * NN + n0 + tx] = tile[ty + i * 8][tx];
    }
}

extern "C" void kernel_launch(void* const* d_in, const int* in_sizes, int n_in,
                              void* d_out, int out_size, void* d_ws, size_t ws_size,
                              hipStream_t stream) {
    (void)in_sizes; (void)n_in; (void)out_size; (void)ws_size;
    const float* p   = (const float*)d_in[0];
    const float* f   = (const float*)d_in[1];
    const int*   idx = (const int*)d_in[2];
    const float* w0  = (const float*)d_in[3];
    const float* g0  = (const float*)d_in[4];
    const float* b0  = (const float*)d_in[5];
    const float* w1  = (const float*)d_in[6];
    const float* g1  = (const float*)d_in[7];
    const float* b1  = (const float*)d_in[8];
    const float* w2  = (const float*)d_in[9];
    const float* g2  = (const float*)d_in[10];
    const float* b2  = (const float*)d_in[11];
    float* out = (float*)d_out;

    float* ws     = (float*)d_ws;
    float* ft     = ws;
    float* hmaxb  = ft     + (size_t)BB * NN * CC;
    float* hminb  = hmaxb  + (size_t)BB * NN * CC;
    float* h1     = hminb  + (size_t)BB * NN * CC;
    float* h2     = h1     + (size_t)BB * NN * MIDC;
    float* stats0 = h2     + (size_t)BB * NN * CC;
    float* stats1 = stats0 + 2 * CC;
    float* stats2 = stats1 + 2 * MIDC;
    float* sb0    = stats2 + 2 * CC;
    float* sb1    = sb0 + 2 * CC;
    float* sb2    = sb1 + 2 * MIDC;

    k_zero<<<3, 256, 0, stream>>>(stats0, 2 * (CC + MIDC + CC));
    k_transpose<<<dim3(NN / 32, CC / 32, BB), dim3(32, 8), 0, stream>>>(f, ft);
    k_stage0<<<BB * NN / 4, 128, 0, stream>>>(p, ft, idx, w0, hmaxb, hminb, stats0);
    k_finalize<<<1, 64, 0, stream>>>(stats0, g0, b0, sb0, 1.f / ((float)BB * NN * KK), CC);
    k_stage1<<<BB * NN / 16, 128, 0, stream>>>(hmaxb, hminb, sb0, w1, h1, stats1);
    k_finalize<<<1, 256, 0, stream>>>(stats1, g1, b1, sb1, 1.f / ((float)BB * NN), MIDC);
    k_stage2<<<BB * NN / 64, 128, 0, stream>>>(h1, sb1, w2, h2, stats2);
    k_finalize<<<1, 64, 0, stream>>>(stats2, g2, b2, sb2, 1.f / ((float)BB * NN), CC);
    k_out<<<dim3(NN / 32, CC / 32, BB), dim3(32, 8), 0, stream>>>(h2, ft, sb2, out);
}